// LightGCNConv_32719060860987
// MI455X (gfx1250) — compile-verified
//
#include <hip/hip_runtime.h>
#include <stdint.h>

#define DIM 64
#define TILE 512      // edges staged per tile (512 * 8B * 2 arrays * 2 buffers = 16KB LDS)
#define THREADS 256   // 8 waves (wave32)

// ---------------------------------------------------------------------------
// CDNA5 async DMA: 16B global -> LDS, tracked on ASYNCcnt (per-lane addresses)
// ---------------------------------------------------------------------------
__device__ __forceinline__ void async_load16(uint32_t lds_off, const void* gptr) {
  asm volatile("global_load_async_to_lds_b128 %0, %1, off"
               :: "v"(lds_off), "v"((uint64_t)(uintptr_t)gptr)
               : "memory");
}

// ---------------------------------------------------------------------------
// Main scatter kernel (placed FIRST so the disasm snippet shows it).
// Double-buffered async staging of (row,col) tiles into LDS; one wave per
// edge; lane l owns dims [2l, 2l+1].
// ---------------------------------------------------------------------------
__global__ __launch_bounds__(THREADS) void scatter_kernel(
    const float2* __restrict__ x2,
    const long long* __restrict__ rows,
    const long long* __restrict__ cols,
    const float* __restrict__ dis,
    float* __restrict__ out,
    long long E, int num_tiles) {
  __shared__ long long s_rows[2][TILE];
  __shared__ long long s_cols[2][TILE];

  const int tid  = threadIdx.x;
  const int lane = tid & 31;
  const int wv   = tid >> 5;
  const int nb   = gridDim.x;

  // Per-thread LDS byte offsets for this thread's 16B staging slots (hoisted).
  const uint32_t lr0 = (uint32_t)(uintptr_t)&s_rows[0][tid * 2];
  const uint32_t lr1 = (uint32_t)(uintptr_t)&s_rows[1][tid * 2];
  const uint32_t lc0 = (uint32_t)(uintptr_t)&s_cols[0][tid * 2];
  const uint32_t lc1 = (uint32_t)(uintptr_t)&s_cols[1][tid * 2];

  // Each thread DMAs 2 row-indices and 2 col-indices (16B each) per tile.
  auto stage = [&](int t, int b) {
    long long e0 = (long long)t * TILE + tid * 2;
    if (e0 > E - 2) e0 = E - 2;   // clamp: duplicated loads land in never-read slots
    if (e0 < 0) e0 = 0;
    async_load16(b ? lr1 : lr0, rows + e0);
    async_load16(b ? lc1 : lc0, cols + e0);
  };

  int t = blockIdx.x;
  if (t < num_tiles) stage(t, 0);   // prologue: ASYNCcnt = 2 per wave
  int buf = 0;

  for (; t < num_tiles; t += nb) {
    int tn = t + nb;
    if (tn < num_tiles) {
      stage(tn, buf ^ 1);                                   // cnt -> 4
      asm volatile("s_wait_asynccnt 0x2" ::: "memory");     // current tile landed (in-order)
    } else {
      asm volatile("s_wait_asynccnt 0x0" ::: "memory");     // last tile: drain
    }
    __syncthreads();   // publish other waves' DMA results

    long long ebase = (long long)t * TILE;
    #pragma unroll 2
    for (int e = wv; e < TILE; e += THREADS / 32) {
      if (ebase + e < E) {   // wave-uniform guard, EXEC stays full
        // broadcast LDS read + scalarize: row/col/norm live in SGPRs
        int r = __builtin_amdgcn_readfirstlane((int)s_rows[buf][e]);
        int c = __builtin_amdgcn_readfirstlane((int)s_cols[buf][e]);
        float nrm = dis[r] * dis[c];
        // coalesced 256B gather of x[r]; 2 dims per lane
        float2 v = x2[(size_t)r * (DIM / 2) + lane];
        float* o = out + (size_t)c * DIM + lane * 2;
        atomicAdd(o,     v.x * nrm);   // global_atomic_add_f32, L2-resident
        atomicAdd(o + 1, v.y * nrm);
      }
    }
    __syncthreads();   // everyone done reading `buf` before it is re-staged
    buf ^= 1;
  }
}

// ---------------------------------------------------------------------------
// Zero-fill (float4 vectorized, grid-stride)
// ---------------------------------------------------------------------------
__global__ __launch_bounds__(THREADS) void zero_f4_kernel(float* __restrict__ p, long long n) {
  long long n4 = n >> 2;
  long long i = (long long)blockIdx.x * blockDim.x + threadIdx.x;
  long long stride = (long long)gridDim.x * blockDim.x;
  float4 z = make_float4(0.f, 0.f, 0.f, 0.f);
  for (; i < n4; i += stride) ((float4*)p)[i] = z;
  if (blockIdx.x == 0 && threadIdx.x == 0)
    for (long long k = n4 << 2; k < n; ++k) p[k] = 0.f;
}

// ---------------------------------------------------------------------------
// Degree histogram via f32 atomics (counts < 2^24, exact in f32)
// ---------------------------------------------------------------------------
__global__ __launch_bounds__(THREADS) void degree_kernel(const long long* __restrict__ rows,
                                                         float* __restrict__ deg, long long E) {
  long long i = (long long)blockIdx.x * blockDim.x + threadIdx.x;
  long long stride = (long long)gridDim.x * blockDim.x;
  for (; i < E; i += stride)
    atomicAdd(&deg[(int)rows[i]], 1.0f);
}

// ---------------------------------------------------------------------------
// deg -> deg^{-1/2} in place, isolated nodes -> 0
// ---------------------------------------------------------------------------
__global__ __launch_bounds__(THREADS) void rsqrt_kernel(float* __restrict__ deg, int N) {
  int i = blockIdx.x * blockDim.x + threadIdx.x;
  if (i < N) {
    float d = deg[i];
    deg[i] = d > 0.f ? rsqrtf(d) : 0.f;
  }
}

// ---------------------------------------------------------------------------
// Host launcher
// ---------------------------------------------------------------------------
extern "C" void kernel_launch(void* const* d_in, const int* in_sizes, int n_in,
                              void* d_out, int out_size, void* d_ws, size_t ws_size,
                              hipStream_t stream) {
  const float*     x  = (const float*)d_in[0];
  const long long* ei = (const long long*)d_in[1];   // int64 per reference

  const long long N = (long long)in_sizes[0] / DIM;
  const long long E = (long long)in_sizes[1] / 2;
  const long long* rows = ei;        // edge_index[0]
  const long long* cols = ei + E;    // edge_index[1]

  float* out = (float*)d_out;
  float* deg = (float*)d_ws;         // N floats: degree, then deg^{-1/2} in place

  if (N <= 0) return;

  // 1) zero deg and out (harness poisons buffers; we own initialization)
  long long outN = N * (long long)DIM;
  long long zb1 = (N / 4) / THREADS + 1;      if (zb1 > 2048) zb1 = 2048;
  long long zb2 = (outN / 4) / THREADS + 1;   if (zb2 > 8192) zb2 = 8192;
  zero_f4_kernel<<<(int)zb1, THREADS, 0, stream>>>(deg, N);
  zero_f4_kernel<<<(int)zb2, THREADS, 0, stream>>>(out, outN);

  if (E <= 0) return;

  // 2) degree histogram
  long long db = (E + THREADS - 1) / THREADS; if (db > 8192) db = 8192;
  degree_kernel<<<(int)db, THREADS, 0, stream>>>(rows, deg, E);

  // 3) deg^{-1/2}
  int rb = (int)((N + THREADS - 1) / THREADS);
  rsqrt_kernel<<<rb, THREADS, 0, stream>>>(deg, (int)N);

  // 4) gather-scale-scatter with async-DMA double buffering
  int num_tiles = (int)((E + TILE - 1) / TILE);
  int sb = num_tiles < 2048 ? num_tiles : 2048;
  scatter_kernel<<<sb, THREADS, 0, stream>>>((const float2*)x, rows, cols, deg, out,
                                             E, num_tiles);
}